// SAConv_52742198395235
// MI455X (gfx1250) — compile-verified
//
#include <hip/hip_runtime.h>

typedef __attribute__((ext_vector_type(2))) float v2f;
typedef __attribute__((ext_vector_type(8))) float v8f;

#define B_   8
#define C_   64
#define H_   56
#define W_   56
#define G_   8
#define D_   8
#define KS_  7
#define PAD_ 3
#define TS   16
#define PS   (TS + KS_ - 1)     /* 22  padded tile edge            */
#define NPP  (PS * PS)          /* 484 padded tile pixels          */
#define NPPP 496                /* padded to 31 x 16 WMMA columns  */
#define NKV  (NPPP / 16)        /* 31 column tiles                 */
#define NTILE 4                 /* ceil(56/16)                     */

__global__ __launch_bounds__(256) void saconv_fused(
    const float* __restrict__ x,
    const float* __restrict__ wq,
    const float* __restrict__ wk,
    const float* __restrict__ wv,
    float* __restrict__ out)
{
    // channel-pair interleaved: elem(ch, p) = arr[(ch>>1)*NPIX + p].{x|y}
    __shared__ float  sW[3][16][16];          // q/k/v weights, 8x8 in 16x16 block
    __shared__ float2 sX[4 * NPPP];           // padded x patch (8 ch as 4 pairs)
    __shared__ float2 sK[4 * NPPP];           // projected K
    __shared__ float2 sV[4 * NPPP];           // projected V
    __shared__ float2 sQ[4 * TS * TS];        // projected Q (inner tile)

    const int tid  = threadIdx.x;
    const int tile = blockIdx.x;
    const int g    = blockIdx.y;
    const int b    = blockIdx.z;
    const int ty = tile / NTILE, tx = tile % NTILE;
    const int y0 = ty * TS,      x0 = tx * TS;

    // ---- 1) weight blocks -> LDS (zero-extended 8x8 -> 16x16) ----
    for (int i = tid; i < 3 * 256; i += 256) {
        int t = i >> 8, e = i & 255;
        int m = e >> 4, k = e & 15;
        const float* wp = (t == 0) ? wq : ((t == 1) ? wk : wv);
        sW[t][m][k] = (m < 8 && k < 8) ? wp[g * 64 + m * 8 + k] : 0.0f;
    }

    // ---- 2) zero-padded x patch -> LDS (incl. zeroed 12-px tail) ----
    {
        const float* xb = x + ((size_t)b * C_ + (size_t)g * D_) * (H_ * W_);
        for (int p = tid; p < NPPP; p += 256) {
            int py = p / PS, px = p % PS;
            int gy = y0 - PAD_ + py, gx = x0 - PAD_ + px;
            bool ok = (p < NPP) & (gy >= 0) & (gy < H_) & (gx >= 0) & (gx < W_);
            int base = (gy * W_ + gx);
            #pragma unroll
            for (int pr = 0; pr < 4; ++pr) {
                float2 t;
                t.x = ok ? xb[(2 * pr)     * (H_ * W_) + base] : 0.0f;
                t.y = ok ? xb[(2 * pr + 1) * (H_ * W_) + base] : 0.0f;
                sX[pr * NPPP + p] = t;
            }
        }
    }
    __syncthreads();

    // ---- 3) q/k/v projections via V_WMMA_F32_16X16X4_F32 ----
    {
        // force wave id into an SGPR: job loops become scalar branches
        const int wave = __builtin_amdgcn_readfirstlane(tid >> 5);
        const int lane = tid & 31;
        const int n    = lane & 15;       // M (A) / N (B,D) index
        const int hi   = lane >> 4;       // 0: K pair {0,1}; 1: K pair {2,3}
        const bool lo16 = (lane < 16);

        // ---- K projection (A hoisted: loop-invariant weights) ----
        {
            v2f a0, a1;
            a0.x = sW[1][n][hi * 2];     a0.y = sW[1][n][hi * 2 + 1];
            a1.x = sW[1][n][4 + hi * 2]; a1.y = sW[1][n][4 + hi * 2 + 1];
            for (int j = wave; j < NKV; j += 8) {
                int p = j * 16 + n;
                float2 b0 = sX[hi * NPPP + p];
                float2 b1 = sX[(2 + hi) * NPPP + p];
                v2f bm0; bm0.x = b0.x; bm0.y = b0.y;
                v2f bm1; bm1.x = b1.x; bm1.y = b1.y;
                v8f c = {0.f,0.f,0.f,0.f,0.f,0.f,0.f,0.f};
                c = __builtin_amdgcn_wmma_f32_16x16x4_f32(false, a0, false, bm0, (short)0, c, false, false);
                c = __builtin_amdgcn_wmma_f32_16x16x4_f32(false, a1, false, bm1, (short)0, c, false, false);
                if (lo16) {
                    #pragma unroll
                    for (int pr = 0; pr < 4; ++pr) {
                        float2 t; t.x = c[2 * pr]; t.y = c[2 * pr + 1];
                        sK[pr * NPPP + p] = t;
                    }
                }
            }
        }
        // ---- V projection ----
        {
            v2f a0, a1;
            a0.x = sW[2][n][hi * 2];     a0.y = sW[2][n][hi * 2 + 1];
            a1.x = sW[2][n][4 + hi * 2]; a1.y = sW[2][n][4 + hi * 2 + 1];
            for (int j = wave; j < NKV; j += 8) {
                int p = j * 16 + n;
                float2 b0 = sX[hi * NPPP + p];
                float2 b1 = sX[(2 + hi) * NPPP + p];
                v2f bm0; bm0.x = b0.x; bm0.y = b0.y;
                v2f bm1; bm1.x = b1.x; bm1.y = b1.y;
                v8f c = {0.f,0.f,0.f,0.f,0.f,0.f,0.f,0.f};
                c = __builtin_amdgcn_wmma_f32_16x16x4_f32(false, a0, false, bm0, (short)0, c, false, false);
                c = __builtin_amdgcn_wmma_f32_16x16x4_f32(false, a1, false, bm1, (short)0, c, false, false);
                if (lo16) {
                    #pragma unroll
                    for (int pr = 0; pr < 4; ++pr) {
                        float2 t; t.x = c[2 * pr]; t.y = c[2 * pr + 1];
                        sV[pr * NPPP + p] = t;
                    }
                }
            }
        }
        // ---- Q projection (inner 16 rows, contiguous 16-px spans in patch) ----
        {
            v2f a0, a1;
            a0.x = sW[0][n][hi * 2];     a0.y = sW[0][n][hi * 2 + 1];
            a1.x = sW[0][n][4 + hi * 2]; a1.y = sW[0][n][4 + hi * 2 + 1];
            for (int iy = wave; iy < TS; iy += 8) {
                int p = (iy + PAD_) * PS + PAD_ + n;
                float2 b0 = sX[hi * NPPP + p];
                float2 b1 = sX[(2 + hi) * NPPP + p];
                v2f bm0; bm0.x = b0.x; bm0.y = b0.y;
                v2f bm1; bm1.x = b1.x; bm1.y = b1.y;
                v8f c = {0.f,0.f,0.f,0.f,0.f,0.f,0.f,0.f};
                c = __builtin_amdgcn_wmma_f32_16x16x4_f32(false, a0, false, bm0, (short)0, c, false, false);
                c = __builtin_amdgcn_wmma_f32_16x16x4_f32(false, a1, false, bm1, (short)0, c, false, false);
                if (lo16) {
                    int qi = iy * 16 + n;
                    #pragma unroll
                    for (int pr = 0; pr < 4; ++pr) {
                        float2 t; t.x = c[2 * pr]; t.y = c[2 * pr + 1];
                        sQ[pr * (TS * TS) + qi] = t;
                    }
                }
            }
        }
    }
    __syncthreads();

    // ---- 4) 7x7 neighborhood attention, online softmax (one pixel / lane) ----
    {
        const int iy = tid >> 4, ix = tid & 15;
        float2 qv[4], acc[4];
        #pragma unroll
        for (int pr = 0; pr < 4; ++pr) {
            qv[pr] = sQ[pr * (TS * TS) + tid];
            acc[pr].x = 0.f; acc[pr].y = 0.f;
        }
        float m = -3.0e38f, ssum = 0.f;

        for (int ki = 0; ki < KS_; ++ki) {
            #pragma unroll
            for (int kj = 0; kj < KS_; ++kj) {
                int p = (iy + ki) * PS + (ix + kj);   // < NPP always
                float l = 0.f;
                #pragma unroll
                for (int pr = 0; pr < 4; ++pr) {
                    float2 kk = sK[pr * NPPP + p];    // ds_load_b64
                    l = fmaf(qv[pr].x, kk.x, l);
                    l = fmaf(qv[pr].y, kk.y, l);
                }
                float mn   = fmaxf(m, l);
                float corr = __expf(m - mn);
                float e    = __expf(l - mn);
                ssum = ssum * corr + e;
                #pragma unroll
                for (int pr = 0; pr < 4; ++pr) {
                    float2 vv = sV[pr * NPPP + p];    // ds_load_b64
                    acc[pr].x = fmaf(acc[pr].x, corr, e * vv.x);
                    acc[pr].y = fmaf(acc[pr].y, corr, e * vv.y);
                }
                m = mn;
            }
        }

        int gy = y0 + iy, gx = x0 + ix;
        if (gy < H_ && gx < W_) {
            float inv = 1.0f / ssum;
            float* ob = out + (((size_t)b * C_ + (size_t)g * D_) * H_ + gy) * W_ + gx;
            #pragma unroll
            for (int pr = 0; pr < 4; ++pr) {
                ob[(size_t)(2 * pr)     * (H_ * W_)] = acc[pr].x * inv;
                ob[(size_t)(2 * pr + 1) * (H_ * W_)] = acc[pr].y * inv;
            }
        }
    }
}

extern "C" void kernel_launch(void* const* d_in, const int* in_sizes, int n_in,
                              void* d_out, int out_size, void* d_ws, size_t ws_size,
                              hipStream_t stream) {
    const float* x  = (const float*)d_in[0];
    // d_in[1] = r  (unused, encoding='none')
    const float* wq = (const float*)d_in[2];
    const float* wk = (const float*)d_in[3];
    const float* wv = (const float*)d_in[4];
    float* out = (float*)d_out;

    dim3 grid(NTILE * NTILE, G_, B_);   // 16 tiles x 8 heads x 8 batch = 1024 WGs
    saconv_fused<<<grid, dim3(256), 0, stream>>>(x, wq, wk, wv, out);
}